// VRNNLoss_18897856102820
// MI455X (gfx1250) — compile-verified
//
#include <hip/hip_runtime.h>

// ---------------------------------------------------------------------------
// VRNN loss: fused single-pass streaming reduction over 8x [64,4096,128] f32
// tensors (1 GB read, scalar out). Memory-bound: 23.3 TB/s -> ~43us floor.
// Wave32, v4f (b128) non-temporal loads, deterministic 2-stage reduction.
// Cross-lane reduction implemented with V_WMMA_F32_16X16X4_F32 (B = ones).
// ---------------------------------------------------------------------------

#define T_DIM 64
#define B_DIM 4096
#define D_DIM 128
#define ROWS (T_DIM * B_DIM)          // 262144 rows of D=128
#define NBLOCKS 2048
#define TPB 256                        // 8 waves (wave32)
#define WAVES_PER_BLOCK (TPB / 32)
#define TOTAL_WAVES (NBLOCKS * WAVES_PER_BLOCK)   // 16384 -> 16 rows/wave
#define HALF_LOG2PI 0.9189385332046727f

typedef float v4f __attribute__((ext_vector_type(4)));
typedef float v2f __attribute__((ext_vector_type(2)));
typedef float v8f __attribute__((ext_vector_type(8)));

__device__ __forceinline__ v4f ntload4(const float* __restrict__ p) {
    // streaming data, zero reuse: non-temporal to avoid L2 rinse
    return __builtin_nontemporal_load((const v4f*)p);
}

__device__ __forceinline__ v4f expv4(v4f v) {
    v4f r;
    r.x = __expf(v.x);
    r.y = __expf(v.y);
    r.z = __expf(v.z);
    r.w = __expf(v.w);
    return r;
}

// Sum one float per lane across the wave using V_WMMA_F32_16X16X4_F32.
// A layout (f32 16x4): lanes 0-15 -> A[m][0] (VGPR0), A[m][1] (VGPR1);
//                      lanes 16-31 -> A[m][2], A[m][3].
// With A[m][0]=p[m], A[m][2]=p[m+16], K=1/3 zeroed, B=ones:
//   D[m][n] = p[m] + p[m+16]  (replicated over n).
// D layout: VGPR j = row j (lanes 0-15) / row j+8 (lanes 16-31), so summing
// the 8 D VGPRs per lane gives sum(rows 0..7) in lanes<16 and sum(rows 8..15)
// in lanes>=16; one xor-16 shuffle add completes the 32-lane total.
__device__ __forceinline__ float wave_reduce_wmma(float v) {
    v2f a; a.x = v;    a.y = 0.0f;
    v2f b; b.x = 1.0f; b.y = 0.0f;
    v8f c = {};
    v8f d = __builtin_amdgcn_wmma_f32_16x16x4_f32(
        /*neg_a=*/false, a, /*neg_b=*/false, b,
        /*c_mod=*/(short)0, c, /*reuse_a=*/false, /*reuse_b=*/false);
    float s = d[0] + d[1] + d[2] + d[3] + d[4] + d[5] + d[6] + d[7];
    s += __shfl_xor(s, 16, 32);
    return s;
}

__global__ void __launch_bounds__(TPB)
vrnn_loss_stage1(const float* __restrict__ pm,   // all_prior_mean [T,B,D]
                 const float* __restrict__ ps,   // all_prior_std
                 const float* __restrict__ xx,   // all_x
                 const float* __restrict__ em,   // all_enc_mean
                 const float* __restrict__ es,   // all_enc_std
                 const float* __restrict__ dm,   // all_dec_mean
                 const float* __restrict__ dsd,  // all_dec_std
                 const float* __restrict__ msk,  // msk [B,T,D]
                 float* __restrict__ ws) {
    const int lane = threadIdx.x & 31;
    const int wave = threadIdx.x >> 5;
    const int gwave = blockIdx.x * WAVES_PER_BLOCK + wave;
    const int d0 = lane * 4;

    v4f acc4 = {};

    // ROWS (262144) is divisible by TOTAL_WAVES (16384): 16 rows/wave,
    // uniform trip count -> EXEC stays all-ones for the WMMA reduction.
    for (int r = gwave; r < ROWS; r += TOTAL_WAVES) {
        const int t = r >> 12;            // r / B_DIM
        const int b = r & (B_DIM - 1);    // r % B_DIM
        const size_t ai = (size_t)r * D_DIM + d0;                    // [t,b,d]
        const size_t mi = ((size_t)b * T_DIM + t) * D_DIM + d0;      // [b,t,d]

        const v4f vpm = ntload4(pm + ai);
        const v4f vps = ntload4(ps + ai);
        const v4f vx  = ntload4(xx + ai);
        const v4f vem = ntload4(em + ai);
        const v4f ves = ntload4(es + ai);
        const v4f vdm = ntload4(dm + ai);
        const v4f vds = ntload4(dsd + ai);
        const v4f vm  = ntload4(msk + mi);

        // KL(enc || prior), log-variance parameterization
        const v4f dmean = vem - vpm;
        const v4f kld = vps - ves + (expv4(ves) + dmean * dmean) * expv4(-vps)
                        - 1.0f;

        // masked diagonal-Gaussian NLL term
        const v4f logstd = 0.5f * vds * vm;
        const v4f z = (vx - vdm) * vm * expv4(-logstd);

        acc4 += 0.5f * kld + (HALF_LOG2PI + logstd + 0.5f * z * z);
    }

    const float acc = (acc4.x + acc4.y) + (acc4.z + acc4.w);
    const float wsum = wave_reduce_wmma(acc);

    __shared__ float smem[WAVES_PER_BLOCK];
    if (lane == 0) smem[wave] = wsum;
    __syncthreads();
    if (threadIdx.x == 0) {
        float s = 0.0f;
        #pragma unroll
        for (int i = 0; i < WAVES_PER_BLOCK; ++i) s += smem[i];
        ws[blockIdx.x] = s;   // every element of ws[0..NBLOCKS) written per call
    }
}

__global__ void __launch_bounds__(32)
vrnn_loss_stage2(const float* __restrict__ ws, float* __restrict__ out) {
    float acc = 0.0f;
    // NBLOCKS divisible by 32 -> uniform loop, deterministic order.
    for (int i = threadIdx.x; i < NBLOCKS; i += 32) acc += ws[i];
    const float tot = wave_reduce_wmma(acc);
    if (threadIdx.x == 0) out[0] = tot * (1.0f / (float)B_DIM);
}

extern "C" void kernel_launch(void* const* d_in, const int* in_sizes, int n_in,
                              void* d_out, int out_size, void* d_ws, size_t ws_size,
                              hipStream_t stream) {
    const float* pm  = (const float*)d_in[0];  // all_prior_mean
    const float* ps  = (const float*)d_in[1];  // all_prior_std
    const float* xx  = (const float*)d_in[2];  // all_x
    const float* em  = (const float*)d_in[3];  // all_enc_mean
    const float* es  = (const float*)d_in[4];  // all_enc_std
    const float* dm  = (const float*)d_in[5];  // all_dec_mean
    const float* dsd = (const float*)d_in[6];  // all_dec_std
    const float* msk = (const float*)d_in[7];  // msk
    // d_in[8] (eval_x) and d_in[9] (eval_msk) are unused by the reference.

    float* ws = (float*)d_ws;                  // NBLOCKS partials (8 KB)
    float* out = (float*)d_out;

    vrnn_loss_stage1<<<NBLOCKS, TPB, 0, stream>>>(pm, ps, xx, em, es, dm, dsd,
                                                  msk, ws);
    vrnn_loss_stage2<<<1, 32, 0, stream>>>(ws, out);
}